// Emu3RVQ_11166914969805
// MI455X (gfx1250) — compile-verified
//
#include <hip/hip_runtime.h>

typedef __attribute__((ext_vector_type(2))) float v2f;
typedef __attribute__((ext_vector_type(8))) float v8f;

#define K_CB   32768
#define NQ     4
#define N_ROWS 8192
#define HW     1024   // 32*32

__global__ void rvq_zero_loss(float* loss) {
    if (threadIdx.x == 0) *loss = 0.0f;
}

__global__ __launch_bounds__(128) void rvq_main(
        const float* __restrict__ x,        // [2,4,4,32,32] = [bt(8), c(4), hw(1024)]
        const float* __restrict__ cb,       // [4, 32768, 4]
        float* __restrict__ zq_out,         // same layout as x
        float* __restrict__ loss_out,       // scalar
        float* __restrict__ codes_out) {    // [8192] (stored as float)
    const int lane = threadIdx.x & 31;
    const int wave = threadIdx.x >> 5;
    const int l16  = lane & 15;
    const bool hi  = lane >= 16;            // holds comps 2,3
    const int c0   = hi ? 2 : 0;

    const int row = (blockIdx.x * 4 + wave) * 16 + l16;   // this lane's row (M = l16)
    const int bt  = row >> 10;
    const int hw  = row & 1023;

    // Load this lane's two residual components (A-fragment layout for 16x4 f32 WMMA)
    const float* xr = x + (size_t)(bt * 4 + c0) * HW + hw;
    float rx = xr[0];
    float ry = xr[HW];
    float zx = 0.0f, zy = 0.0f;
    float lsq = 0.0f;
    int   code0 = 0;

    __shared__ int lds_idx[4 * 16];         // per-wave winning-index slots
    int* slot = lds_idx + wave * 16;

    for (int q = 0; q < NQ; ++q) {
        const float* cbq = cb + (size_t)q * K_CB * 4;

        v2f a; a.x = rx; a.y = ry;          // A: residual tile (16 rows x 4)
        float best[8];
        int   bidx[8];
        #pragma unroll
        for (int v = 0; v < 8; ++v) { best[v] = -3.402823466e38f; bidx[v] = 0; }

        #pragma unroll 4
        for (int base = 0; base < K_CB; base += 16) {
            const int n = base + l16;
            // B: 16 codebook entries; combined half-waves read 256 contiguous bytes
            v2f b = *(const v2f*)(cbq + (size_t)n * 4 + c0);
            // cv = 0.5*||e_n||^2 via cross-half shuffle (comps 0,1 <-> 2,3)
            float ss = b.x * b.x + b.y * b.y;
            float cv = 0.5f * (ss + __shfl_xor(ss, 16, 32));
            // D = A x B (C folds to inline 0) -> 16x16 dot products r_m . e_n
            v8f d = __builtin_amdgcn_wmma_f32_16x16x4_f32(
                false, a, false, b, (short)0, (v8f){}, false, false);
            #pragma unroll
            for (int v = 0; v < 8; ++v) {
                float s = d[v] - cv;        // argmax(r.e - 0.5||e||^2) == argmin dist
                if (s > best[v]) { best[v] = s; bidx[v] = n; }
            }
        }

        // Reduce across the 16 lanes of each half (N-dim); keep lowest idx on ties
        #pragma unroll
        for (int m = 1; m < 16; m <<= 1) {
            #pragma unroll
            for (int v = 0; v < 8; ++v) {
                float ob = __shfl_xor(best[v], m, 32);
                int   oi = __shfl_xor(bidx[v], m, 32);
                if (ob > best[v] || (ob == best[v] && oi < bidx[v])) {
                    best[v] = ob; bidx[v] = oi;
                }
            }
        }

        // Publish winners: lane 0 -> rows 0..7, lane 16 -> rows 8..15 (DS in-order per wave)
        if (l16 == 0) {
            #pragma unroll
            for (int v = 0; v < 8; ++v) slot[(hi ? 8 : 0) + v] = bidx[v];
        }
        int myIdx = slot[l16];

        // Gather quantized pair, update loss / z_q / residual
        const float* e = cbq + (size_t)myIdx * 4 + c0;
        float qx = e[0], qy = e[1];
        float dx = rx - qx, dy = ry - qy;
        lsq += dx * dx + dy * dy;
        zx += qx; zy += qy;
        rx -= qx; ry -= qy;
        if (q == 0) code0 = myIdx;
    }

    // Store z_q in [bt, c, hw] layout
    float* zo = zq_out + (size_t)(bt * 4 + c0) * HW + hw;
    zo[0]  = zx;
    zo[HW] = zy;
    if (!hi) codes_out[row] = (float)code0;

    // Wave-reduce squared-error sum, one atomic per wave
    #pragma unroll
    for (int m = 1; m < 32; m <<= 1) lsq += __shfl_xor(lsq, m, 32);
    if (lane == 0) atomicAdd(loss_out, lsq * (1.25f / (float)(N_ROWS * 4)));
}

extern "C" void kernel_launch(void* const* d_in, const int* in_sizes, int n_in,
                              void* d_out, int out_size, void* d_ws, size_t ws_size,
                              hipStream_t stream) {
    const float* x  = (const float*)d_in[0];   // [2,4,4,32,32]
    const float* cb = (const float*)d_in[1];   // [4,32768,4]
    float* out = (float*)d_out;
    float* zq    = out;                 // 32768 floats
    float* loss  = out + 32768;         // 1 float
    float* codes = out + 32769;         // 8192 floats

    rvq_zero_loss<<<1, 1, 0, stream>>>(loss);
    rvq_main<<<128, 128, 0, stream>>>(x, cb, zq, loss, codes);
}